// GraphSpectralFilterLayer_41059887350004
// MI455X (gfx1250) — compile-verified
//
#include <hip/hip_runtime.h>
#include <hip/hip_bf16.h>
#include <cstdint>

typedef __attribute__((ext_vector_type(2))) float v2f;
typedef __attribute__((ext_vector_type(8))) float v8f;

#define IN_F   256
#define OUT_F  32

// ---------------------------------------------------------------------------
// Kernel 1: h = X @ W via V_WMMA_F32_16X16X4_F32.
// One wave computes a 16-row x 32-col tile of h (two 16x16 accumulators).
//
// W is staged in LDS in PAIR-PACKED form:
//   sWp[kp*32 + n] = { W[2*kp][n], W[2*kp+1][n] }   (kp = k/2)
// so each lane's B operand (two consecutive-K values for one column) is a
// single aligned ds_load_b64 into an even VGPR pair — no v_mov repacking.
//
// f32 16x16x4 VGPR layouts (ISA 7.12.2):
//   A (16x4):  lane m=lane%16, kb=(lane/16)*2 holds A[m][k+kb], A[m][k+kb+1]
//   B (4x16):  lane n=lane%16, kb=(lane/16)*2 holds B[k+kb][n], B[k+kb+1][n]
//   C/D 16x16: VGPR r, lanes 0-15 -> (M=r, N=lane); lanes 16-31 -> (M=r+8)
// ---------------------------------------------------------------------------
__global__ void __launch_bounds__(256)
gemm_h_wmma(const float* __restrict__ X, const float* __restrict__ W,
            float* __restrict__ H, int N) {
    __shared__ v2f sWp[(IN_F / 2) * OUT_F];          // 128*32 pairs = 32 KB
    for (int i = threadIdx.x; i < (IN_F / 2) * OUT_F; i += blockDim.x) {
        const int kp = i >> 5;                       // K-pair index
        const int n  = i & 31;                       // column
        v2f p;
        p.x = W[(2 * kp)     * OUT_F + n];
        p.y = W[(2 * kp + 1) * OUT_F + n];
        sWp[i] = p;
    }
    __syncthreads();

    const int lane = threadIdx.x & 31;
    const int wave = threadIdx.x >> 5;
    const int wavesPerBlock = blockDim.x >> 5;
    const int rowBlock = blockIdx.x * wavesPerBlock + wave;
    const int row0 = rowBlock * 16;
    if (row0 >= N) return;                           // wave-uniform: EXEC stays all-1s

    const int m   = lane & 15;
    const int kb  = (lane >> 4) << 1;                // 0 or 2
    const int kpb = lane >> 4;                       // pair offset: 0 or 1

    const float* __restrict__ xrow = X + (size_t)(row0 + m) * IN_F;

    v8f acc0 = {};
    v8f acc1 = {};
    #pragma unroll 4
    for (int k = 0; k < IN_F; k += 4) {
        const int kp = (k >> 1) + kpb;               // this lane's K-pair
        v2f a  = *(const v2f*)(xrow + k + kb);       // A[m][k+kb..k+kb+1]
        v2f b0 = sWp[kp * OUT_F + m];                // B[.][m]      (cols 0-15)
        v2f b1 = sWp[kp * OUT_F + 16 + m];           // B[.][m+16]   (cols 16-31)
        acc0 = __builtin_amdgcn_wmma_f32_16x16x4_f32(false, a, false, b0,
                                                     (short)0, acc0, false, false);
        acc1 = __builtin_amdgcn_wmma_f32_16x16x4_f32(false, a, false, b1,
                                                     (short)0, acc1, false, false);
    }

    // Write back: VGPR r holds rows (row0 + mOff + r), col = lane%16 (+16 for acc1)
    const int n    = lane & 15;
    const int mOff = (lane >> 4) * 8;
    float* __restrict__ hBase = H + (size_t)(row0 + mOff) * OUT_F + n;
    if (row0 + 16 <= N) {                            // full tile: no per-row guards
        #pragma unroll
        for (int r = 0; r < 8; ++r) {
            hBase[(size_t)r * OUT_F]      = acc0[r];
            hBase[(size_t)r * OUT_F + 16] = acc1[r];
        }
    } else {
        #pragma unroll
        for (int r = 0; r < 8; ++r) {
            if (row0 + mOff + r < N) {
                hBase[(size_t)r * OUT_F]      = acc0[r];
                hBase[(size_t)r * OUT_F + 16] = acc1[r];
            }
        }
    }
}

// ---------------------------------------------------------------------------
// Kernel 2: per-order edge scatter.
// One wave per edge-chunk; the 32 lanes cover the 32 output features, so the
// h[col][:] gather and the num[row][:] atomic adds are fully coalesced.
// ---------------------------------------------------------------------------
__global__ void __launch_bounds__(256)
edge_scatter(const float* __restrict__ coeff,        // [E] for this order
             const long long* __restrict__ rowIdx,   // [E]
             const long long* __restrict__ colIdx,   // [E]
             const float* __restrict__ H,            // [N*32]
             float* __restrict__ num,                // [N*32], pre-zeroed
             float* __restrict__ divisor,            // [N],    pre-zeroed
             int E) {
    const int lane   = threadIdx.x & 31;
    const int wave   = (blockIdx.x * blockDim.x + threadIdx.x) >> 5;
    const int nWaves = (gridDim.x * blockDim.x) >> 5;

    for (int e = wave; e < E; e += nWaves) {
        const float c = coeff[e];
        // leakyrelu keeps negatives negative -> masked to NEG_FILL; NaN fails
        // c>=0 -> NEG_FILL; exp(NEG_FILL)=0; positives: min(exp(c), 9e15).
        float v = (c >= 0.0f) ? c : -9e15f;
        v = fminf(__expf(v), 9e15f);

        const long long r  = rowIdx[e];
        const long long cc = colIdx[e];
        const float hv = H[(size_t)cc * OUT_F + lane];

        unsafeAtomicAdd(&num[(size_t)r * OUT_F + lane], v * hv);
        if (lane == 0)
            unsafeAtomicAdd(&divisor[r], v);
    }
}

// ---------------------------------------------------------------------------
// Kernel 3: h_prime = elu(num / max(divisor,!=0)) -> out[:, o*32:(o+1)*32]
// ---------------------------------------------------------------------------
__global__ void __launch_bounds__(256)
finalize(const float* __restrict__ num, const float* __restrict__ divisor,
         float* __restrict__ out, int N, int order, int orderTot) {
    const int idx = blockIdx.x * blockDim.x + threadIdx.x;   // n*32 + j
    if (idx >= N * OUT_F) return;
    const int n = idx >> 5;
    const int j = idx & 31;
    float d = divisor[n];
    d = (d == 0.0f) ? 1.0f : d;
    const float x = num[idx] / d;
    const float y = (x > 0.0f) ? x : (__expf(x) - 1.0f);     // elu
    out[(size_t)n * (orderTot * OUT_F) + order * OUT_F + j] = y;
}

// ---------------------------------------------------------------------------
// Launch
// ---------------------------------------------------------------------------
extern "C" void kernel_launch(void* const* d_in, const int* in_sizes, int n_in,
                              void* d_out, int out_size, void* d_ws, size_t ws_size,
                              hipStream_t stream) {
    const float*     X     = (const float*)d_in[0];      // [N, 256]
    const float*     W     = (const float*)d_in[1];      // [256, 32]
    const float*     coeff = (const float*)d_in[2];      // [ORDER, E]
    const long long* eidx  = (const long long*)d_in[3];  // [2, E] int64

    const int N     = in_sizes[0] / IN_F;
    const int E     = in_sizes[3] / 2;
    const int ORDER = in_sizes[2] / E;

    const long long* rowIdx = eidx;
    const long long* colIdx = eidx + E;

    // Workspace layout: H | num | div   (num+div zeroed per order, contiguous)
    float* H       = (float*)d_ws;                     // N*32
    float* num     = H + (size_t)N * OUT_F;            // N*32
    float* divisor = num + (size_t)N * OUT_F;          // N
    const size_t zeroBytes = ((size_t)N * OUT_F + (size_t)N) * sizeof(float);

    // 1) h = X @ W  (WMMA)
    {
        const int wavesPerBlock = 8;                   // 256 threads
        const int rowBlocks = (N + 15) / 16;
        const int grid = (rowBlocks + wavesPerBlock - 1) / wavesPerBlock;
        gemm_h_wmma<<<grid, 256, 0, stream>>>(X, W, H, N);
    }

    // 2) per spectral order: zero accumulators, scatter edges, finalize
    const int edgeBlocks = 4096;                       // 32768 waves over 1.6M edges
    const int finBlocks  = (N * OUT_F + 255) / 256;
    for (int o = 0; o < ORDER; ++o) {
        hipMemsetAsync(num, 0, zeroBytes, stream);
        edge_scatter<<<edgeBlocks, 256, 0, stream>>>(
            coeff + (size_t)o * E, rowIdx, colIdx, H, num, divisor, E);
        finalize<<<finBlocks, 256, 0, stream>>>(
            num, divisor, (float*)d_out, N, o, ORDER);
    }
}